// DualBranchGNN_75127567941761
// MI455X (gfx1250) — compile-verified
//
#include <hip/hip_runtime.h>
#include <math.h>

// ---------------------------------------------------------------------------
// DualBranchGNN for gfx1250 (MI455X). fp32 WMMA (v_wmma_f32_16x16x4_f32) for
// the three dense GEMMs; edge phase is scatter/gather with float atomics.
// Memory-bound problem: minimize HBM passes, vectorize to b128 where possible.
// ---------------------------------------------------------------------------

typedef __attribute__((ext_vector_type(2))) float v2f;
typedef __attribute__((ext_vector_type(8))) float v8f;

#define SIGMA_F 30.0f

static __device__ __forceinline__ float leaky02(float a) {
    return a > 0.0f ? a : 0.2f * a;
}

static __device__ __forceinline__ float elu1(float a) {
    return a > 0.0f ? a : (expf(a) - 1.0f);
}

// monotonic float atomic-max via sign-split integer atomics (portable)
static __device__ __forceinline__ void atomic_max_float(float* addr, float val) {
    int iv = __float_as_int(val);
    if (iv >= 0) {
        atomicMax((int*)addr, iv);
    } else {
        atomicMin((unsigned int*)addr, (unsigned int)iv);
    }
}

// ---------------------------------------------------------------------------
// K1: two node-feature GEMMs  O[N,64] = X[N,64] @ W^T  (W is [64,64] row-major)
// One wave computes a full 16x64 output tile: 4 f32-WMMA accumulators, A frag
// reused across the 4 N-tiles, 64 v_wmma_f32_16x16x4_f32 per wave.
// For the GAT matrix the wave also computes as = xp.att_src, ad = xp.att_dst
// in-register (half-wave shfl_xor reduction), saving a full re-read of xp.
// f32 A frag (16x4): lanes 0-15 -> K=k0+0,k0+1 ; lanes 16-31 -> K=k0+2,k0+3.
// C/D 16x16 f32: VGPR r holds row r (lanes 0-15) / row r+8 (lanes 16-31).
// ---------------------------------------------------------------------------
__global__ void gemm_node_kernel(const float* __restrict__ Xa, const float* __restrict__ Wa,
                                 float* __restrict__ Oa,
                                 const float* __restrict__ Xb, const float* __restrict__ Wb,
                                 float* __restrict__ Ob,
                                 const float* __restrict__ att_src,
                                 const float* __restrict__ att_dst,
                                 float* __restrict__ asv, float* __restrict__ adv,
                                 int N, int tilesM) {
    int gwave = (blockIdx.x * blockDim.x + threadIdx.x) >> 5;
    int lane  = threadIdx.x & 31;
    if (gwave >= 2 * tilesM) return;

    bool isA = gwave < tilesM;
    int mt = isA ? gwave : (gwave - tilesM);
    const float* X = isA ? Xa : Xb;
    const float* W = isA ? Wa : Wb;
    float*       O = isA ? Oa : Ob;

    int m0 = mt * 16;
    int row = m0 + (lane & 15);
    if (row >= N) row = N - 1;                 // clamp (tail tile); stores guarded
    int koff  = (lane & 16) ? 2 : 0;
    int nlane = lane & 15;

    const float* xr = X + (size_t)row * 64 + koff;

    v8f acc[4] = {{}, {}, {}, {}};
#pragma unroll
    for (int k0 = 0; k0 < 64; k0 += 4) {
        v2f a;
        a.x = xr[k0]; a.y = xr[k0 + 1];
#pragma unroll
        for (int t = 0; t < 4; ++t) {
            const float* wr = W + (size_t)(t * 16 + nlane) * 64 + koff;
            v2f b;
            b.x = wr[k0]; b.y = wr[k0 + 1];
            acc[t] = __builtin_amdgcn_wmma_f32_16x16x4_f32(false, a, false, b, (short)0,
                                                           acc[t], false, false);
        }
    }

    int mb = m0 + ((lane & 16) ? 8 : 0);
#pragma unroll
    for (int t = 0; t < 4; ++t) {
        int col = t * 16 + nlane;
#pragma unroll
        for (int r = 0; r < 8; ++r) {
            int m = mb + r;
            if (m < N) O[(size_t)m * 64 + col] = acc[t][r];
        }
    }

    if (isA) {
        // attention dots computed from the registers we already hold
        float ps[8] = {0, 0, 0, 0, 0, 0, 0, 0};
        float pd[8] = {0, 0, 0, 0, 0, 0, 0, 0};
#pragma unroll
        for (int t = 0; t < 4; ++t) {
            int n = t * 16 + nlane;
            float ws_ = att_src[n];
            float wd_ = att_dst[n];
#pragma unroll
            for (int r = 0; r < 8; ++r) {
                ps[r] += acc[t][r] * ws_;
                pd[r] += acc[t][r] * wd_;
            }
        }
#pragma unroll
        for (int off = 1; off < 16; off <<= 1) {
#pragma unroll
            for (int r = 0; r < 8; ++r) {
                ps[r] += __shfl_xor(ps[r], off, 32);
                pd[r] += __shfl_xor(pd[r], off, 32);
            }
        }
        if (nlane == 0) {
#pragma unroll
            for (int r = 0; r < 8; ++r) {
                int m = mb + r;
                if (m < N) { asv[m] = ps[r]; adv[m] = pd[r]; }
            }
        }
    }
}

// K2: init segment state. Self loop is always valid -> amax starts at a_self.
__global__ void node_init_kernel(const float* __restrict__ asv, const float* __restrict__ adv,
                                 float* __restrict__ amax, float* __restrict__ degv, int N) {
    int i = blockIdx.x * blockDim.x + threadIdx.x;
    if (i >= N) return;
    amax[i] = leaky02(asv[i] + adv[i]);
    degv[i] = 1.0f;  // self-loop GCN weight
}

// K3: per-edge weight MLP + gaussian + mask; attention logit; seg-max / deg.
__global__ void edge_weight_kernel(const int* __restrict__ src, const int* __restrict__ dst,
                                   const float* __restrict__ noise, const float* __restrict__ coord,
                                   const float* __restrict__ fc1_w, const float* __restrict__ fc1_b,
                                   const float* __restrict__ fc2_w, const float* __restrict__ fc2_b,
                                   const float* __restrict__ asv, const float* __restrict__ adv,
                                   float* __restrict__ wE, float* __restrict__ aE,
                                   float* __restrict__ amax, float* __restrict__ degv, int E) {
    int e = blockIdx.x * blockDim.x + threadIdx.x;
    if (e >= E) return;
    int s = src[e], d = dst[e];

    float nf[10];
#pragma unroll
    for (int i = 0; i < 10; ++i) nf[i] = noise[s * 10 + i];
    float ni = fc2_b[0];
#pragma unroll
    for (int j = 0; j < 10; ++j) {
        float h = fc1_b[j];
#pragma unroll
        for (int i = 0; i < 10; ++i) h += fc1_w[j * 10 + i] * nf[i];
        ni += fc2_w[j] * elu1(h);
    }

    float dx = coord[s * 2 + 0] - coord[d * 2 + 0];
    float dy = coord[s * 2 + 1] - coord[d * 2 + 1];
    float gw = expf(-(dx * dx + dy * dy) / (2.0f * SIGMA_F * SIGMA_F));
    float x  = gw * (1.0f + ni);
    float sg = 1.0f / (1.0f + expf(-(x - 1.0f)));
    float w  = 0.1f + 1.9f * sg;
    wE[e] = w;

    float a = leaky02(asv[s] + adv[d]);
    aE[e] = a;
    if (w >= 0.2f) {
        atomic_max_float(&amax[d], a);
        atomicAdd(&degv[d], w);
    }
}

// K4: esum starts with self-loop exp; deg -> dinv (overwrite in place).
__global__ void node_stats_kernel(const float* __restrict__ asv, const float* __restrict__ adv,
                                  const float* __restrict__ amax,
                                  float* __restrict__ degv, float* __restrict__ esum, int N) {
    int i = blockIdx.x * blockDim.x + threadIdx.x;
    if (i >= N) return;
    float a = leaky02(asv[i] + adv[i]);
    esum[i] = expf(a - amax[i]);
    float dg = degv[i];
    degv[i] = (dg > 0.0f) ? rsqrtf(fmaxf(dg, 1e-12f)) : 0.0f;
}

// K5: per-edge exp + seg-sum (valid edges only; invalid -> e = 0).
__global__ void edge_exp_kernel(const int* __restrict__ dst,
                                const float* __restrict__ wE, float* __restrict__ aE,
                                const float* __restrict__ amax, float* __restrict__ esum, int E) {
    int e = blockIdx.x * blockDim.x + threadIdx.x;
    if (e >= E) return;
    if (wE[e] >= 0.2f) {
        int d = dst[e];
        float ev = expf(aE[e] - amax[d]);
        aE[e] = ev;
        atomicAdd(&esum[d], ev);
    } else {
        aE[e] = 0.0f;
    }
}

// K6: self-loop contributions written (not added) into cat = d_out+N, [N][128].
// One thread per (node, 4-channel group); b128 loads/stores.
__global__ void node_self_kernel(const float* __restrict__ xp, const float* __restrict__ xg,
                                 const float* __restrict__ asv, const float* __restrict__ adv,
                                 const float* __restrict__ amax, const float* __restrict__ dinv,
                                 const float* __restrict__ esum,
                                 float* __restrict__ cat, int total) {
    int t = blockIdx.x * blockDim.x + threadIdx.x;
    if (t >= total) return;
    int i = t >> 4, c0 = (t & 15) * 4;
    float a = leaky02(asv[i] + adv[i]);
    float alpha = expf(a - amax[i]) / (esum[i] + 1e-16f);
    float dv = dinv[i];
    float dd = dv * dv;

    float4 xv = *(const float4*)(xp + (size_t)i * 64 + c0);
    float4 gv = *(const float4*)(xg + (size_t)i * 64 + c0);
    float4 ol = make_float4(alpha * xv.x, alpha * xv.y, alpha * xv.z, alpha * xv.w);
    float4 og = make_float4(dd * gv.x, dd * gv.y, dd * gv.z, dd * gv.w);
    *(float4*)(cat + (size_t)i * 128 + c0)      = ol;
    *(float4*)(cat + (size_t)i * 128 + 64 + c0) = og;
}

// K7: edge scatter — one thread per (edge, 8-channel group): per-edge scalars
// loaded once per group, gathers as global_load_b128, 16 float atomics.
__global__ void edge_scatter_kernel(const int* __restrict__ src, const int* __restrict__ dst,
                                    const float* __restrict__ wE, const float* __restrict__ eV,
                                    const float* __restrict__ esum, const float* __restrict__ dinv,
                                    const float* __restrict__ xp, const float* __restrict__ xg,
                                    float* __restrict__ cat, int total) {
    int t = blockIdx.x * blockDim.x + threadIdx.x;
    if (t >= total) return;
    int e = t >> 3, c0 = (t & 7) * 8;
    float w = wE[e];
    if (w < 0.2f) return;
    int s = src[e], d = dst[e];
    float alpha = eV[e] / (esum[d] + 1e-16f);
    float nrm   = dinv[s] * w * dinv[d];

    const float4* xs = (const float4*)(xp + (size_t)s * 64 + c0);
    const float4* gs = (const float4*)(xg + (size_t)s * 64 + c0);
    float4 x0 = xs[0], x1 = xs[1];
    float4 g0 = gs[0], g1 = gs[1];

    float* cl = cat + (size_t)d * 128 + c0;
    float* cg = cl + 64;
    atomicAdd(&cl[0], alpha * x0.x);
    atomicAdd(&cl[1], alpha * x0.y);
    atomicAdd(&cl[2], alpha * x0.z);
    atomicAdd(&cl[3], alpha * x0.w);
    atomicAdd(&cl[4], alpha * x1.x);
    atomicAdd(&cl[5], alpha * x1.y);
    atomicAdd(&cl[6], alpha * x1.z);
    atomicAdd(&cl[7], alpha * x1.w);
    atomicAdd(&cg[0], nrm * g0.x);
    atomicAdd(&cg[1], nrm * g0.y);
    atomicAdd(&cg[2], nrm * g0.z);
    atomicAdd(&cg[3], nrm * g0.w);
    atomicAdd(&cg[4], nrm * g1.x);
    atomicAdd(&cg[5], nrm * g1.y);
    atomicAdd(&cg[6], nrm * g1.z);
    atomicAdd(&cg[7], nrm * g1.w);
}

// K8: bias + ELU (GAT half) / ReLU (GCN half), in place, b128 per thread.
__global__ void node_act_kernel(const float* __restrict__ gat_b, const float* __restrict__ gcn_b,
                                float* __restrict__ cat, int total) {
    int t = blockIdx.x * blockDim.x + threadIdx.x;
    if (t >= total) return;
    int i = t >> 4, c0 = (t & 15) * 4;
    float* cl = cat + (size_t)i * 128 + c0;
    float* cg = cl + 64;
    float4 v = *(float4*)cl;
    float4 b = *(const float4*)(gat_b + c0);
    v.x = elu1(v.x + b.x); v.y = elu1(v.y + b.y);
    v.z = elu1(v.z + b.z); v.w = elu1(v.w + b.w);
    *(float4*)cl = v;
    float4 g  = *(float4*)cg;
    float4 bg = *(const float4*)(gcn_b + c0);
    g.x = fmaxf(g.x + bg.x, 0.0f); g.y = fmaxf(g.y + bg.y, 0.0f);
    g.z = fmaxf(g.z + bg.z, 0.0f); g.w = fmaxf(g.w + bg.w, 0.0f);
    *(float4*)cg = g;
}

// K9: fuse MLP. One wave per 16-node tile: hf[16,64] via WMMA over K=128,
// then ReLU + dot with fuse_w2 using a half-wave shfl_xor reduction -> pred.
__global__ void fuse_kernel(const float* __restrict__ cat,
                            const float* __restrict__ w1, const float* __restrict__ b1,
                            const float* __restrict__ w2, const float* __restrict__ b2,
                            float* __restrict__ pred, int N, int tilesM) {
    int gwave = (blockIdx.x * blockDim.x + threadIdx.x) >> 5;
    int lane  = threadIdx.x & 31;
    if (gwave >= tilesM) return;

    int m0 = gwave * 16;
    int row = m0 + (lane & 15);
    if (row >= N) row = N - 1;
    int koff  = (lane & 16) ? 2 : 0;
    int nlane = lane & 15;

    const float* cr = cat + (size_t)row * 128 + koff;
    v8f acc[4] = {{}, {}, {}, {}};

#pragma unroll 4
    for (int k0 = 0; k0 < 128; k0 += 4) {
        v2f a;
        a.x = cr[k0]; a.y = cr[k0 + 1];
#pragma unroll
        for (int t = 0; t < 4; ++t) {
            const float* wr = w1 + (size_t)(t * 16 + nlane) * 128 + koff;
            v2f b;
            b.x = wr[k0]; b.y = wr[k0 + 1];
            acc[t] = __builtin_amdgcn_wmma_f32_16x16x4_f32(false, a, false, b, (short)0,
                                                           acc[t], false, false);
        }
    }

    float part[8] = {0, 0, 0, 0, 0, 0, 0, 0};
#pragma unroll
    for (int t = 0; t < 4; ++t) {
        int n = t * 16 + nlane;
        float bb = b1[n];
        float ww = w2[n];
#pragma unroll
        for (int r = 0; r < 8; ++r) {
            float h = fmaxf(acc[t][r] + bb, 0.0f);   // ReLU(hf)
            part[r] += h * ww;
        }
    }
    // sum over the 16 lanes of each half-wave (rows 0-7 in lanes 0-15, 8-15 in 16-31)
#pragma unroll
    for (int off = 1; off < 16; off <<= 1) {
#pragma unroll
        for (int r = 0; r < 8; ++r) part[r] += __shfl_xor(part[r], off, 32);
    }
    if (nlane == 0) {
        int mb = m0 + ((lane & 16) ? 8 : 0);
        float bias = b2[0];
#pragma unroll
        for (int r = 0; r < 8; ++r) {
            int m = mb + r;
            if (m < N) pred[m] = part[r] + bias;
        }
    }
}

// ---------------------------------------------------------------------------
extern "C" void kernel_launch(void* const* d_in, const int* in_sizes, int n_in,
                              void* d_out, int out_size, void* d_ws, size_t ws_size,
                              hipStream_t stream) {
    const float* x_local  = (const float*)d_in[0];
    const float* x_global = (const float*)d_in[1];
    const float* noise    = (const float*)d_in[2];
    const float* coord    = (const float*)d_in[3];
    const int*   eidx     = (const int*)d_in[4];
    const float* fc1_w    = (const float*)d_in[5];
    const float* fc1_b    = (const float*)d_in[6];
    const float* fc2_w    = (const float*)d_in[7];
    const float* fc2_b    = (const float*)d_in[8];
    const float* gat_w    = (const float*)d_in[9];
    const float* gat_b    = (const float*)d_in[10];
    const float* att_src  = (const float*)d_in[11];
    const float* att_dst  = (const float*)d_in[12];
    const float* gcn_w    = (const float*)d_in[13];
    const float* gcn_b    = (const float*)d_in[14];
    const float* fuse_w1  = (const float*)d_in[15];
    const float* fuse_b1  = (const float*)d_in[16];
    const float* fuse_w2  = (const float*)d_in[17];
    const float* fuse_b2  = (const float*)d_in[18];

    const int N = in_sizes[0] / 64;
    const int E = in_sizes[4] / 2;
    const int* src = eidx;
    const int* dst = eidx + E;

    // workspace layout (floats)
    float* ws   = (float*)d_ws;
    float* xp   = ws;                              // N*64
    float* xg   = xp + (size_t)N * 64;             // N*64
    float* asv  = xg + (size_t)N * 64;             // N
    float* adv  = asv + N;                         // N
    float* amax = adv + N;                         // N
    float* degv = amax + N;                        // N (later holds dinv)
    float* esum = degv + N;                        // N
    float* wE   = esum + N;                        // E
    float* aE   = wE + E;                          // E (logit, then exp value)

    float* pred = (float*)d_out;                   // N
    float* cat  = pred + N;                        // N*128

    const int tilesM = (N + 15) / 16;
    const int BLK = 256;

    // 1. node GEMMs (+ fused attention dots): xp, xg, asv, adv
    {
        int threads = 2 * tilesM * 32;
        gemm_node_kernel<<<(threads + BLK - 1) / BLK, BLK, 0, stream>>>(
            x_local, gat_w, xp, x_global, gcn_w, xg,
            att_src, att_dst, asv, adv, N, tilesM);
    }
    // 2. segment state init (amax = self-loop logit, deg = 1)
    node_init_kernel<<<(N + BLK - 1) / BLK, BLK, 0, stream>>>(asv, adv, amax, degv, N);
    // 3. edge weights + logits + seg-max/deg
    edge_weight_kernel<<<(E + BLK - 1) / BLK, BLK, 0, stream>>>(
        src, dst, noise, coord, fc1_w, fc1_b, fc2_w, fc2_b, asv, adv, wE, aE, amax, degv, E);
    // 4. esum init (self term) + dinv
    node_stats_kernel<<<(N + BLK - 1) / BLK, BLK, 0, stream>>>(asv, adv, amax, degv, esum, N);
    // 5. edge exp + seg-sum
    edge_exp_kernel<<<(E + BLK - 1) / BLK, BLK, 0, stream>>>(dst, wE, aE, amax, esum, E);
    // 6. self-loop contributions -> cat (full overwrite, no zero-init needed)
    {
        int total = N * 16;
        node_self_kernel<<<(total + BLK - 1) / BLK, BLK, 0, stream>>>(
            xp, xg, asv, adv, amax, degv, esum, cat, total);
    }
    // 7. edge scatter (GAT + GCN aggregation), b128 gathers
    {
        int total = E * 8;
        edge_scatter_kernel<<<(total + BLK - 1) / BLK, BLK, 0, stream>>>(
            src, dst, wE, aE, esum, degv, xp, xg, cat, total);
    }
    // 8. bias + activations
    {
        int total = N * 16;
        node_act_kernel<<<(total + BLK - 1) / BLK, BLK, 0, stream>>>(gat_b, gcn_b, cat, total);
    }
    // 9. fuse MLP -> pred
    {
        int threads = tilesM * 32;
        fuse_kernel<<<(threads + BLK - 1) / BLK, BLK, 0, stream>>>(
            cat, fuse_w1, fuse_b1, fuse_w2, fuse_b2, pred, N, tilesM);
    }
}